// TokenRoutedExpertMLP_27943057228227
// MI455X (gfx1250) — compile-verified
//
#include <hip/hip_runtime.h>
#include <math.h>
#include <stdint.h>

// ---------------------------------------------------------------------------
// Token-routed expert MLP (MoE), fp32, CDNA5 (gfx1250, wave32).
//   TOKENS=131072, HIDDEN=768, INTER=192, EXPERTS=4
// Plan:
//   1) bucket tokens by expert (count -> scan -> scatter) into d_ws
//   2) fused MLP GEMM kernel: per block = 16 tokens of one expert,
//      V_WMMA_F32_16X16X4_F32 for both GEMMs; activation tile staged into
//      LDS with GLOBAL_LOAD_ASYNC_TO_LDS_B128 (ASYNCcnt) — CDNA5 async path.
// ---------------------------------------------------------------------------

typedef float v2f __attribute__((ext_vector_type(2)));
typedef float v8f __attribute__((ext_vector_type(8)));

#define T_TOKENS 131072
#define HID      768
#define FF       192
#define NF2      384      // 2*FF
#define NEXP     4
#define MTILE    16
#define A_PITCH  388      // 384 cols + 4 pad -> 16 rows hit distinct LDS banks
#define ACT_PITCH 196     // 192 + 4 pad

// ws int layout: [0..3]=counts  [4..7]=cursors  [8..11]=offsets  [16..16+T)=idx

__device__ __forceinline__ v8f wmma_f32_16x16x4(v2f a, v2f b, v8f c) {
    // (neg_a, A, neg_b, B, c_mod, C, reuse_a, reuse_b)
    return __builtin_amdgcn_wmma_f32_16x16x4_f32(false, a, false, b,
                                                 (short)0, c, false, false);
}

// Async memory->LDS copy of 16 bytes (per lane), tracked by ASYNCcnt.
__device__ __forceinline__ void async_copy_b128(unsigned lds_byte_addr,
                                                const void* gptr) {
    unsigned long long ga = (unsigned long long)(uintptr_t)gptr;
    asm volatile("global_load_async_to_lds_b128 %0, %1, off"
                 :: "v"(lds_byte_addr), "v"(ga)
                 : "memory");
}
__device__ __forceinline__ void async_wait0() {
    asm volatile("s_wait_asynccnt 0x0" ::: "memory");
}

__global__ void moe_zero_kernel(int* wsi) {
    if (threadIdx.x < 16) wsi[threadIdx.x] = 0;
}

__global__ void moe_count_kernel(const int* __restrict__ eid, int* wsi) {
    int t = blockIdx.x * 256 + threadIdx.x;
    if (t < T_TOKENS) atomicAdd(&wsi[eid[t]], 1);
}

__global__ void moe_scan_kernel(int* wsi) {
    if (threadIdx.x == 0) {
        int o = 0;
        for (int i = 0; i < NEXP; ++i) {
            int c = wsi[i];
            wsi[8 + i] = o;   // offsets
            wsi[4 + i] = o;   // cursors
            o += c;
        }
    }
}

__global__ void moe_scatter_kernel(const int* __restrict__ eid, int* wsi) {
    int t = blockIdx.x * 256 + threadIdx.x;
    if (t < T_TOKENS) {
        int e = eid[t];
        int p = atomicAdd(&wsi[4 + e], 1);
        wsi[16 + p] = t;
    }
}

__global__ __launch_bounds__(256)
void moe_mlp_kernel(const float* __restrict__ hidden,
                    const float* __restrict__ gate_up,
                    const float* __restrict__ down,
                    const int*   __restrict__ wsi,
                    float*       __restrict__ out)
{
    // smem is Ah (16 x 388) during GEMM1, then ACT (16 x 196) for GEMM2.
    __shared__ __align__(16) float smem[MTILE * A_PITCH];
    __shared__ __align__(16) float GU[MTILE * A_PITCH];
    __shared__ int   s_tok[MTILE];

    const int* counts  = wsi + 0;
    const int* offsets = wsi + 8;
    const int* idx     = wsi + 16;

    // ---- map blockIdx.x -> (expert e, tile tl within expert) ----
    int b = blockIdx.x;
    int e = -1, tl = 0, cnt = 0, acc_t = 0;
    for (int i = 0; i < NEXP; ++i) {
        int c  = counts[i];
        int nt = (c + MTILE - 1) / MTILE;
        if (e < 0 && b < acc_t + nt) { e = i; tl = b - acc_t; cnt = c; }
        acc_t += nt;
    }
    if (e < 0) return;                       // uniform per block

    const int off  = offsets[e];
    const int row0 = tl * MTILE;
    int rem = cnt - row0;
    const int rowsValid = rem < MTILE ? rem : MTILE;

    const int tid  = threadIdx.x;
    const int wave = tid >> 5;
    const int lane = tid & 31;
    const int ln   = lane & 15;
    const int hi   = lane >> 4;              // half-wave select
    const int sel  = hi * 2;                 // K sub-offset for A/B fragments

    if (tid < MTILE) {
        int r = row0 + tid;
        if (r > cnt - 1) r = cnt - 1;        // clamp (cnt>=1 when tiles exist)
        s_tok[tid] = idx[off + r];
    }
    __syncthreads();

    // ------------------- GEMM1: gu[16x384] = A[16x768] * W[768x384] ----------
    const int colbase = wave * 48;           // 3 N-tiles per wave
    v8f acc0 = {}, acc1 = {}, acc2v = {};

    const int row = tid >> 4;                // 16 threads per row for staging
    const int lc  = tid & 15;
    const unsigned lds_row_base =
        (unsigned)(uintptr_t)(&smem[0]) + (unsigned)(row * A_PITCH * 4);

    for (int phase = 0; phase < 2; ++phase) {
        if (phase) __syncthreads();          // all reads of smem done
        // stage 16 rows x 384 cols of gathered hidden straight into LDS
        // via async b128 copies (no VGPR round-trip, ASYNCcnt tracked)
        {
            const float* srcf = hidden + (size_t)s_tok[row] * HID + phase * NF2;
            #pragma unroll
            for (int j = 0; j < 6; ++j) {    // 96 float4 per row / 16 threads
                int f4 = lc + 16 * j;
                async_copy_b128(lds_row_base + (unsigned)(f4 * 16),
                                srcf + f4 * 4);
            }
            async_wait0();                   // ASYNCcnt==0: LDS writes landed
        }
        __syncthreads();

        const int kglob0 = phase * NF2;
        #pragma unroll 4
        for (int k = 0; k < NF2; k += 4) {
            v2f a = *(const v2f*)&smem[ln * A_PITCH + k + sel];
            const float* bp =
                gate_up + ((size_t)(e * HID + kglob0 + k + sel)) * NF2 + colbase + ln;
            v2f b0; b0.x = bp[0];      b0.y = bp[NF2];
            v2f b1; b1.x = bp[16];     b1.y = bp[NF2 + 16];
            v2f b2; b2.x = bp[32];     b2.y = bp[NF2 + 32];
            acc0  = wmma_f32_16x16x4(a, b0, acc0);
            acc1  = wmma_f32_16x16x4(a, b1, acc1);
            acc2v = wmma_f32_16x16x4(a, b2, acc2v);
        }
    }

    // write gu fragments to LDS: VGPR v -> row M = v + 8*hi, col N = base + ln
    #pragma unroll
    for (int v = 0; v < 8; ++v) {
        int m = v + 8 * hi;
        GU[m * A_PITCH + colbase +  0 + ln] = acc0[v];
        GU[m * A_PITCH + colbase + 16 + ln] = acc1[v];
        GU[m * A_PITCH + colbase + 32 + ln] = acc2v[v];
    }
    __syncthreads();

    // ------------------- act = silu(gate) * up  (reuse smem as ACT) ----------
    #pragma unroll
    for (int j = 0; j < 12; ++j) {           // 16*192 / 256 threads
        int i = tid + 256 * j;
        int m = i / FF, f = i - m * FF;
        float g = GU[m * A_PITCH + f];
        float u = GU[m * A_PITCH + FF + f];
        float s = g / (1.0f + __expf(-g));   // silu
        smem[m * ACT_PITCH + f] = s * u;
    }
    __syncthreads();

    // ------------------- GEMM2: out[16x768] = act[16x192] * W[192x768] -------
    const int colbase2 = wave * 96;          // 6 N-tiles per wave
    v8f o0 = {}, o1 = {}, o2 = {}, o3 = {}, o4 = {}, o5 = {};

    #pragma unroll 4
    for (int k = 0; k < FF; k += 4) {
        v2f a = *(const v2f*)&smem[ln * ACT_PITCH + k + sel];
        const float* bp = down + ((size_t)(e * FF + k + sel)) * HID + colbase2 + ln;
        v2f b0; b0.x = bp[ 0]; b0.y = bp[HID +  0];
        v2f b1; b1.x = bp[16]; b1.y = bp[HID + 16];
        v2f b2; b2.x = bp[32]; b2.y = bp[HID + 32];
        v2f b3; b3.x = bp[48]; b3.y = bp[HID + 48];
        v2f b4; b4.x = bp[64]; b4.y = bp[HID + 64];
        v2f b5; b5.x = bp[80]; b5.y = bp[HID + 80];
        o0 = wmma_f32_16x16x4(a, b0, o0);
        o1 = wmma_f32_16x16x4(a, b1, o1);
        o2 = wmma_f32_16x16x4(a, b2, o2);
        o3 = wmma_f32_16x16x4(a, b3, o3);
        o4 = wmma_f32_16x16x4(a, b4, o4);
        o5 = wmma_f32_16x16x4(a, b5, o5);
    }

    // ------------------- scatter stores (row-masked for partial tiles) -------
    int toks[8];
    #pragma unroll
    for (int v = 0; v < 8; ++v) toks[v] = s_tok[v + 8 * hi];

    #pragma unroll
    for (int v = 0; v < 8; ++v) {
        int m = v + 8 * hi;
        if (m < rowsValid) {
            float* op = out + (size_t)toks[v] * HID + colbase2 + ln;
            op[ 0] = o0[v];
            op[16] = o1[v];
            op[32] = o2[v];
            op[48] = o3[v];
            op[64] = o4[v];
            op[80] = o5[v];
        }
    }
}

extern "C" void kernel_launch(void* const* d_in, const int* in_sizes, int n_in,
                              void* d_out, int out_size, void* d_ws, size_t ws_size,
                              hipStream_t stream) {
    const float* hidden  = (const float*)d_in[0];
    const int*   eids    = (const int*)  d_in[1];
    const float* gate_up = (const float*)d_in[2];
    const float* down    = (const float*)d_in[3];
    float*       out     = (float*)d_out;
    int*         wsi     = (int*)d_ws;

    moe_zero_kernel   <<<1, 64, 0, stream>>>(wsi);
    moe_count_kernel  <<<T_TOKENS / 256, 256, 0, stream>>>(eids, wsi);
    moe_scan_kernel   <<<1, 32, 0, stream>>>(wsi);
    moe_scatter_kernel<<<T_TOKENS / 256, 256, 0, stream>>>(eids, wsi);

    const int nblocks = T_TOKENS / MTILE + NEXP;   // covers per-expert padding
    moe_mlp_kernel<<<nblocks, 256, 0, stream>>>(hidden, gate_up, down, wsi, out);
}